// BudgieCausalDepthwiseConv1d_23124103921701
// MI455X (gfx1250) — compile-verified
//
#include <hip/hip_runtime.h>

// Causal depthwise conv1d, B=4, S=4096, H=2048, K=4, fp32.
// Memory-bound streaming kernel: ~256 MiB of traffic, ~0.5 GFLOP -> the
// target is pure HBM bandwidth (23.3 TB/s => ~11.5 us).
// Strategy: one thread owns a float4 channel-group and a 32-long strip of s.
// Sliding 4-row register window -> each input row loaded ~once (35/32 amp).
// b128 loads, NT b128 stores, gfx1250 global_prefetch for the next tile.
// Mask values are hoisted into registers before the loop so the unrolled body
// has no scalar-memory waits (single up-front kmcnt wait instead of 32).

typedef float v4f __attribute__((ext_vector_type(4)));

constexpr int Bc  = 4;
constexpr int Sc  = 4096;
constexpr int Hc  = 2048;
constexpr int TS  = 32;          // s-positions per thread (register sliding window)
constexpr int TPB = 256;         // 8 wave32 per block
constexpr int CG  = Hc / 4;      // 512 float4 channel-groups

__global__ __launch_bounds__(TPB)
void budgie_causal_dwconv1d_kernel(const float* __restrict__ x,
                                   const float* __restrict__ w,
                                   const float* __restrict__ mask,
                                   float* __restrict__ y)
{
    const int cg = blockIdx.x * TPB + threadIdx.x;   // channel group [0, 512)
    const int s0 = blockIdx.y * TS;                  // tile start in S
    const int b  = blockIdx.z;                       // batch
    const int c  = cg << 2;                          // first channel of group

    // Per-channel taps: weight is [H, 4] row-major, so channel (c+i)'s 4 taps
    // are one contiguous float4. w0..w3 = taps for channels c..c+3.
    const v4f* wp = (const v4f*)(w + (size_t)c * 4);
    const v4f w0 = wp[0], w1 = wp[1], w2 = wp[2], w3 = wp[3];

    const float* xb = x + ((size_t)b * Sc) * Hc + c;
    float*       yb = y + ((size_t)b * Sc) * Hc + c;
    const float* mb = mask + (size_t)b * Sc + s0;

    // Hoist the (wave-uniform) mask strip into registers: adjacent uniform
    // loads merge into wide SMEM fetches with one kmcnt wait total.
    float mv[TS];
#pragma unroll
    for (int t = 0; t < TS; ++t) mv[t] = mb[t];

    // gfx1250 prefetch path: pull the next s-tile's first row for this lane.
    if (s0 + TS < Sc)
        __builtin_prefetch(xb + (size_t)(s0 + TS) * Hc, 0, 3);

    const v4f z = {0.f, 0.f, 0.f, 0.f};
    // Causal left context (zero-padded only for the first tile).
    v4f xm3 = (s0 >= 3) ? *(const v4f*)(xb + (size_t)(s0 - 3) * Hc) : z;
    v4f xm2 = (s0 >= 2) ? *(const v4f*)(xb + (size_t)(s0 - 2) * Hc) : z;
    v4f xm1 = (s0 >= 1) ? *(const v4f*)(xb + (size_t)(s0 - 1) * Hc) : z;

#pragma unroll
    for (int t = 0; t < TS; ++t) {
        const int s = s0 + t;
        const v4f xc = *(const v4f*)(xb + (size_t)s * Hc);  // global_load_b128
        const float m = mv[t];

        // y[s] = w[0]*x[s-3] + w[1]*x[s-2] + w[2]*x[s-1] + w[3]*x[s]
        v4f r;
        r.x = ((w0.x * xm3.x + w0.y * xm2.x) + (w0.z * xm1.x + w0.w * xc.x)) * m;
        r.y = ((w1.x * xm3.y + w1.y * xm2.y) + (w1.z * xm1.y + w1.w * xc.y)) * m;
        r.z = ((w2.x * xm3.z + w2.y * xm2.z) + (w2.z * xm1.z + w2.w * xc.z)) * m;
        r.w = ((w3.x * xm3.w + w3.y * xm2.w) + (w3.z * xm1.w + w3.w * xc.w)) * m;

        // NT temporal hint: output is never re-read, keep it out of L2's way.
        __builtin_nontemporal_store(r, (v4f*)(yb + (size_t)s * Hc));

        xm3 = xm2; xm2 = xm1; xm1 = xc;   // slide the causal window
    }
}

extern "C" void kernel_launch(void* const* d_in, const int* in_sizes, int n_in,
                              void* d_out, int out_size, void* d_ws, size_t ws_size,
                              hipStream_t stream)
{
    const float* x    = (const float*)d_in[0];   // [B, S, H]
    const float* w    = (const float*)d_in[1];   // [H, K=4]
    const float* mask = (const float*)d_in[2];   // [B, S]
    float*       y    = (float*)d_out;           // [B, S, H]

    dim3 block(TPB, 1, 1);
    dim3 grid(CG / TPB, Sc / TS, Bc);            // (2, 128, 4)
    budgie_causal_dwconv1d_kernel<<<grid, block, 0, stream>>>(x, w, mask, y);
}